// PairwiseContextProjector_89386859364535
// MI455X (gfx1250) — compile-verified
//
#include <hip/hip_runtime.h>
#include <hip/hip_bf16.h>

// Problem constants (match reference)
#define BATCH 2
#define CH 64
#define LEN 1024
#define IN_DIM 192   // 3*CH
#define EMB 512
#define SHORT_R 10

typedef __attribute__((ext_vector_type(2))) float v2f;
typedef __attribute__((ext_vector_type(4))) float v4f;
typedef __attribute__((ext_vector_type(8))) float v8f;

// ---------------------------------------------------------------------------
// Stage 1: one wave per (b,c,i) row of x_pairwise. Streaming fp32 reduction:
// total sum + local-window sum; derive short/long/global context and scatter
// into ctx[b, i, 3C] layout ready for the GEMM.
// ---------------------------------------------------------------------------
__global__ __launch_bounds__(256) void ctx_reduce_kernel(
    const float* __restrict__ x, float* __restrict__ ctx) {
  const int lane = threadIdx.x & 31;
  const int row  = (int)((blockIdx.x * blockDim.x + threadIdx.x) >> 5); // 0..B*C*L-1
  const int i  = row & (LEN - 1);
  const int bc = row >> 10;            // b*C + c
  const int c  = bc & (CH - 1);
  const int b  = bc >> 6;

  const v4f* __restrict__ xr4 =
      reinterpret_cast<const v4f*>(x + (size_t)row * LEN);

  const int lo = i - SHORT_R;
  const int hi = i + SHORT_R;

  float tot = 0.0f;
  float loc = 0.0f;
#pragma unroll
  for (int chk = 0; chk < 8; ++chk) {
    const v4f v = __builtin_nontemporal_load(&xr4[chk * 32 + lane]);
    const int j0 = chk * 128 + lane * 4;
    tot += (v.x + v.y) + (v.z + v.w);
    loc += (j0     >= lo && j0     <= hi) ? v.x : 0.0f;
    loc += (j0 + 1 >= lo && j0 + 1 <= hi) ? v.y : 0.0f;
    loc += (j0 + 2 >= lo && j0 + 2 <= hi) ? v.z : 0.0f;
    loc += (j0 + 3 >= lo && j0 + 3 <= hi) ? v.w : 0.0f;
  }

#pragma unroll
  for (int off = 16; off > 0; off >>= 1) {
    tot += __shfl_xor(tot, off, 32);
    loc += __shfl_xor(loc, off, 32);
  }

  if (lane == 0) {
    const int n_local = min(i, SHORT_R) + min(LEN - 1 - i, SHORT_R) + 1;
    const float inv_nl = 1.0f / (float)n_local;
    const float inv_nd = 1.0f / (float)(LEN - n_local);
    float* __restrict__ dst = ctx + ((size_t)(b * LEN + i) * IN_DIM + c);
    dst[0]        = loc * inv_nl;              // short_ctx
    dst[CH]       = (tot - loc) * inv_nd;      // long_ctx
    dst[2 * CH]   = tot * (1.0f / (float)LEN); // global_ctx
  }
}

// ---------------------------------------------------------------------------
// fp32 WMMA GEMM: one wave per 16(M) x 64(N) output block, K in steps of 4.
// A fragment reused across the 4 N-subtiles.
// ---------------------------------------------------------------------------
__global__ __launch_bounds__(256) void gemm1_relu_kernel(
    const float* __restrict__ A,     // ctx [2048, 192]
    const float* __restrict__ W,     // W1  [192, 512]
    const float* __restrict__ bias,  // b1  [512]
    float* __restrict__ H) {         // h   [2048, 512]
  const int lane  = threadIdx.x & 31;
  const int wave  = (int)((blockIdx.x * blockDim.x + threadIdx.x) >> 5);
  const int mTile = wave >> 3;   // 0..127
  const int nBlk  = wave & 7;    // 0..7  (n base = nBlk*64)
  const int ln16  = lane & 15;
  const int kgrp  = (lane >> 4) << 1;  // 0 or 2
  const int m     = mTile * 16 + ln16;
  const int nB    = nBlk * 64 + ln16;

  v8f acc0 = {}, acc1 = {}, acc2 = {}, acc3 = {};
  for (int k = 0; k < IN_DIM; k += 4) {
    const v2f a =
        *reinterpret_cast<const v2f*>(A + (size_t)m * IN_DIM + k + kgrp);
    const float* __restrict__ wp = W + (size_t)(k + kgrp) * EMB;
    v2f f0; f0.x = wp[nB];      f0.y = wp[EMB + nB];
    v2f f1; f1.x = wp[nB + 16]; f1.y = wp[EMB + nB + 16];
    v2f f2; f2.x = wp[nB + 32]; f2.y = wp[EMB + nB + 32];
    v2f f3; f3.x = wp[nB + 48]; f3.y = wp[EMB + nB + 48];
    acc0 = __builtin_amdgcn_wmma_f32_16x16x4_f32(false, a, false, f0, (short)0, acc0, false, false);
    acc1 = __builtin_amdgcn_wmma_f32_16x16x4_f32(false, a, false, f1, (short)0, acc1, false, false);
    acc2 = __builtin_amdgcn_wmma_f32_16x16x4_f32(false, a, false, f2, (short)0, acc2, false, false);
    acc3 = __builtin_amdgcn_wmma_f32_16x16x4_f32(false, a, false, f3, (short)0, acc3, false, false);
  }

  const int mOut = mTile * 16 + ((lane >> 4) << 3);  // +0 or +8 half
  v8f accs[4] = {acc0, acc1, acc2, acc3};
#pragma unroll
  for (int t = 0; t < 4; ++t) {
    const int n = nBlk * 64 + t * 16 + ln16;
    const float bv = bias[n];
#pragma unroll
    for (int r = 0; r < 8; ++r) {
      const float v = accs[t][r] + bv;
      H[(size_t)(mOut + r) * EMB + n] = fmaxf(v, 0.0f);
    }
  }
}

__global__ __launch_bounds__(256) void gemm2_kernel(
    const float* __restrict__ A,     // h  [2048, 512]
    const float* __restrict__ W,     // W2 [512, 512]
    const float* __restrict__ bias,  // b2 [512]
    float* __restrict__ O) {         // out [2048, 512]
  const int lane  = threadIdx.x & 31;
  const int wave  = (int)((blockIdx.x * blockDim.x + threadIdx.x) >> 5);
  const int mTile = wave >> 3;
  const int nBlk  = wave & 7;
  const int ln16  = lane & 15;
  const int kgrp  = (lane >> 4) << 1;
  const int m     = mTile * 16 + ln16;
  const int nB    = nBlk * 64 + ln16;

  v8f acc0 = {}, acc1 = {}, acc2 = {}, acc3 = {};
  for (int k = 0; k < EMB; k += 4) {
    const v2f a =
        *reinterpret_cast<const v2f*>(A + (size_t)m * EMB + k + kgrp);
    const float* __restrict__ wp = W + (size_t)(k + kgrp) * EMB;
    v2f f0; f0.x = wp[nB];      f0.y = wp[EMB + nB];
    v2f f1; f1.x = wp[nB + 16]; f1.y = wp[EMB + nB + 16];
    v2f f2; f2.x = wp[nB + 32]; f2.y = wp[EMB + nB + 32];
    v2f f3; f3.x = wp[nB + 48]; f3.y = wp[EMB + nB + 48];
    acc0 = __builtin_amdgcn_wmma_f32_16x16x4_f32(false, a, false, f0, (short)0, acc0, false, false);
    acc1 = __builtin_amdgcn_wmma_f32_16x16x4_f32(false, a, false, f1, (short)0, acc1, false, false);
    acc2 = __builtin_amdgcn_wmma_f32_16x16x4_f32(false, a, false, f2, (short)0, acc2, false, false);
    acc3 = __builtin_amdgcn_wmma_f32_16x16x4_f32(false, a, false, f3, (short)0, acc3, false, false);
  }

  const int mOut = mTile * 16 + ((lane >> 4) << 3);
  v8f accs[4] = {acc0, acc1, acc2, acc3};
#pragma unroll
  for (int t = 0; t < 4; ++t) {
    const int n = nBlk * 64 + t * 16 + ln16;
    const float bv = bias[n];
#pragma unroll
    for (int r = 0; r < 8; ++r) {
      O[(size_t)(mOut + r) * EMB + n] = accs[t][r] + bv;
    }
  }
}

extern "C" void kernel_launch(void* const* d_in, const int* in_sizes, int n_in,
                              void* d_out, int out_size, void* d_ws, size_t ws_size,
                              hipStream_t stream) {
  const float* x  = (const float*)d_in[0];  // [2, 64, 1024, 1024]
  const float* W1 = (const float*)d_in[1];  // [192, 512]
  const float* b1 = (const float*)d_in[2];  // [512]
  const float* W2 = (const float*)d_in[3];  // [512, 512]
  const float* b2 = (const float*)d_in[4];  // [512]
  float* out = (float*)d_out;               // [2, 1024, 512]

  float* ctx = (float*)d_ws;                          // [2048, 192] = 1.5 MB
  float* h   = ctx + (size_t)BATCH * LEN * IN_DIM;    // [2048, 512] = 4 MB

  // Stage 1: 131072 rows, 8 waves (256 threads) per block -> 16384 blocks.
  const int rows = BATCH * CH * LEN;
  ctx_reduce_kernel<<<rows / 8, 256, 0, stream>>>(x, ctx);

  // Stage 2/3: 128 mTiles * 8 nBlocks = 1024 wave-tasks, 8 waves/block.
  gemm1_relu_kernel<<<128, 256, 0, stream>>>(ctx, W1, b1, h);
  gemm2_kernel<<<128, 256, 0, stream>>>(h, W2, b2, out);
}